// TimeCollapseRNN__40175124087522
// MI455X (gfx1250) — compile-verified
//
#include <hip/hip_runtime.h>
#include <hip/hip_bf16.h>
#include <math.h>

// CDNA5 (gfx1250) wave32 WMMA types
typedef float v2f __attribute__((ext_vector_type(2)));
typedef float v8f __attribute__((ext_vector_type(8)));

#define N_SAMPLES  256
#define N_CHANNELS 32
#define LOOKBACK   64
#define N_ASSETS   512
#define HIDDEN     32
#define CL_STRIDE  (LOOKBACK * N_ASSETS)   // x stride between channels (floats)

// Native gfx1250 V_TANH_F32 (single trans op, no branches).
static __device__ __forceinline__ float fast_tanh(float v) {
    return __builtin_amdgcn_tanhf(v);
}

// out[s,h,a] = mean_l tanh( sum_c W_ih[h,c]*x[s,c,l,a] + b_ih[h] + b_hh[h] )
// One wave per (sample, 16-asset tile). fp32 WMMA 16x16x4, K=32 as 8 chained steps,
// two h-tiles (h=0..15, 16..31) sharing the B fragments.
__global__ __launch_bounds__(256)
void TimeCollapseRNN_wmma_kernel(const float* __restrict__ x,
                                 const float* __restrict__ W_ih,
                                 const float* __restrict__ b_ih,
                                 const float* __restrict__ b_hh,
                                 float* __restrict__ out)
{
    const int lane        = threadIdx.x & 31;
    const int waveInBlock = threadIdx.x >> 5;
    const int w           = blockIdx.x * 8 + waveInBlock;   // 0..8191
    const int s           = w >> 5;                         // 512/16 = 32 a-tiles per sample
    const int a0          = (w & 31) * 16;

    const int laneN  = lane & 15;   // column (B/C/D) or row M (A) within tile
    const int laneHi = lane >> 4;   // 0: lanes 0-15, 1: lanes 16-31

    // ---- A fragments: W_ih (HIDDEN x N_CHANNELS), row-major W[h*32 + c].
    // 16x4 f32 A layout: VGPR0 = {lanes0-15: K=0, lanes16-31: K=2}, VGPR1 = {K=1, K=3}.
    // Tile t covers h = 16t + M; k-chunk k covers c = 4k..4k+3.
    v2f afrag[2][8];
#pragma unroll
    for (int t = 0; t < 2; ++t) {
        const int h = 16 * t + laneN;
#pragma unroll
        for (int k = 0; k < 8; ++k) {
            const int c = 4 * k + 2 * laneHi;
            afrag[t][k].x = W_ih[h * N_CHANNELS + c];
            afrag[t][k].y = W_ih[h * N_CHANNELS + c + 1];
        }
    }

    // ---- Bias in C/D layout (VGPR r: lanes0-15 -> M=r, lanes16-31 -> M=r+8),
    // used as the C operand of the first accumulation chain each timestep.
    v8f bias[2];
#pragma unroll
    for (int t = 0; t < 2; ++t) {
#pragma unroll
        for (int r = 0; r < 8; ++r) {
            const int h = 16 * t + r + 8 * laneHi;
            bias[t][r] = b_ih[h] + b_hh[h];
        }
    }

    v8f acc0 = {};
    v8f acc1 = {};

    // x[s][c][l][a]; this lane's column base
    const float* xs = x + (size_t)s * N_CHANNELS * CL_STRIDE + a0 + laneN;

    for (int l = 0; l < LOOKBACK; ++l) {
        const float* xl = xs + (size_t)l * N_ASSETS;

        // B fragments (4x16 f32): VGPR0 = {lanes0-15: K=0, lanes16-31: K=2}, VGPR1 = {K=1, K=3}
        // k-chunk k covers c = 4k..4k+3 for columns a0..a0+15.
        v2f bfrag[8];
#pragma unroll
        for (int k = 0; k < 8; ++k) {
            const int c0 = 4 * k + 2 * laneHi;
            bfrag[k].x = xl[(size_t)c0 * CL_STRIDE];
            bfrag[k].y = xl[(size_t)(c0 + 1) * CL_STRIDE];
        }

        v8f d0 = bias[0];
        v8f d1 = bias[1];
#pragma unroll
        for (int k = 0; k < 8; ++k) {
            d0 = __builtin_amdgcn_wmma_f32_16x16x4_f32(
                    false, afrag[0][k], false, bfrag[k], (short)0, d0, false, false);
        }
#pragma unroll
        for (int k = 0; k < 8; ++k) {
            d1 = __builtin_amdgcn_wmma_f32_16x16x4_f32(
                    false, afrag[1][k], false, bfrag[k], (short)0, d1, false, false);
        }

#pragma unroll
        for (int r = 0; r < 8; ++r) {
            acc0[r] += fast_tanh(d0[r]);
            acc1[r] += fast_tanh(d1[r]);
        }
    }

    // ---- mean over l and store. out[s,h,a], C/D layout -> h = 16t + r + 8*laneHi.
    const float inv = 1.0f / (float)LOOKBACK;
    float* outp = out + (size_t)s * HIDDEN * N_ASSETS + a0 + laneN;
#pragma unroll
    for (int r = 0; r < 8; ++r) {
        const int h0 = r + 8 * laneHi;
        outp[(size_t)h0 * N_ASSETS]        = acc0[r] * inv;
        outp[(size_t)(16 + h0) * N_ASSETS] = acc1[r] * inv;
    }
}

extern "C" void kernel_launch(void* const* d_in, const int* in_sizes, int n_in,
                              void* d_out, int out_size, void* d_ws, size_t ws_size,
                              hipStream_t stream) {
    // setup_inputs order: x, W_ih, W_hh (unused: multiplied by zero hidden state), b_ih, b_hh
    const float* x    = (const float*)d_in[0];
    const float* W_ih = (const float*)d_in[1];
    const float* b_ih = (const float*)d_in[3];
    const float* b_hh = (const float*)d_in[4];
    float* out = (float*)d_out;

    // 8192 waves total: 256 samples x 32 asset-tiles; 8 waves (256 threads) per block.
    const int n_waves = N_SAMPLES * (N_ASSETS / 16);
    dim3 grid(n_waves / 8), block(256);
    TimeCollapseRNN_wmma_kernel<<<grid, block, 0, stream>>>(x, W_ih, b_ih, b_hh, out);
}